// GraphA2C_35596688949779
// MI455X (gfx1250) — compile-verified
//
#include <hip/hip_runtime.h>

typedef __attribute__((ext_vector_type(2))) float v2f;
typedef __attribute__((ext_vector_type(8))) float v8f;

#define NNODES 50000
#define NEDGES 1600000
#define FIN    256
#define DD     128

// ---------------------------------------------------------------------------
// Dense GEMM via fp32 WMMA:  C[M x 128] = A[M x K] @ W[K x 128] (+bias)(+relu)
// One 16x16 tile per wave, 4 waves/block covering 64 cols, grid.y=2 covers 128.
// A-frag (16x4 f32): lane l -> m = l&15, half = l>>4; v0 <-> K=2*half, v1 <-> K=2*half+1
// B-frag (4x16 f32): lane l -> n = l&15; same K mapping.
// C/D (16x16 f32, 8 VGPRs): vgpr r, lane l -> M = r + 8*(l>>4), N = l&15
// ---------------------------------------------------------------------------
__global__ __launch_bounds__(128)
void gemm_wmma_kernel(const float* __restrict__ A, const float* __restrict__ W,
                      const float* __restrict__ bias, float* __restrict__ C,
                      int M, int K, int doRelu)
{
    const int wave  = threadIdx.x >> 5;
    const int lane  = threadIdx.x & 31;
    const int half  = lane >> 4;
    const int l16   = lane & 15;
    const int tileM = blockIdx.x << 4;
    const int tileN = (blockIdx.y << 6) + (wave << 4);

    const float* Arow = A + (size_t)(tileM + l16) * K + 2 * half;
    const float* Bcol = W + tileN + l16 + (size_t)(2 * half) * DD;

    v8f c = {0.f, 0.f, 0.f, 0.f, 0.f, 0.f, 0.f, 0.f};
    #pragma unroll 4
    for (int kb = 0; kb < K; kb += 4) {
        v2f a, b;
        a.x = Arow[kb];
        a.y = Arow[kb + 1];
        b.x = Bcol[(size_t)kb * DD];
        b.y = Bcol[(size_t)(kb + 1) * DD];
        c = __builtin_amdgcn_wmma_f32_16x16x4_f32(
                /*neg_a=*/false, a, /*neg_b=*/false, b,
                /*c_mod=*/(short)0, c, /*reuse_a=*/false, /*reuse_b=*/false);
    }

    const int col = tileN + l16;
    float bv = bias ? bias[col] : 0.f;
    #pragma unroll
    for (int r = 0; r < 8; ++r) {
        int row = tileM + r + 8 * half;
        float v = c[r] + bv;
        if (doRelu) v = v > 0.f ? v : 0.f;
        C[(size_t)row * DD + col] = v;
    }
}

// ---------------------------------------------------------------------------
// CSR construction: histogram of dst, block scan -> offsets, slot fill.
// ---------------------------------------------------------------------------
__global__ void count_kernel(const int* __restrict__ edges, int* __restrict__ counts)
{
    int i = blockIdx.x * blockDim.x + threadIdx.x;
    if (i < NEDGES) atomicAdd(&counts[edges[2 * i + 1]], 1);
}

__global__ __launch_bounds__(1024)
void scan_kernel(const int* __restrict__ counts, int* __restrict__ offs,
                 int* __restrict__ cursor, int n)
{
    __shared__ int carry;
    __shared__ int buf[1024];
    const int tid = threadIdx.x;
    if (tid == 0) carry = 0;
    __syncthreads();
    for (int base = 0; base < n; base += 1024) {
        int i = base + tid;
        int v = (i < n) ? counts[i] : 0;
        buf[tid] = v;
        __syncthreads();
        for (int s = 1; s < 1024; s <<= 1) {
            int t = (tid >= s) ? buf[tid - s] : 0;
            __syncthreads();
            buf[tid] += t;
            __syncthreads();
        }
        int incl = buf[tid] + carry;
        if (i < n) { offs[i + 1] = incl; cursor[i] = incl - v; }
        __syncthreads();
        if (tid == 1023) carry = incl;
        __syncthreads();
    }
    if (tid == 0) offs[0] = 0;
}

__global__ void dis_kernel(const int* __restrict__ counts, float* __restrict__ dis, int n)
{
    int i = blockIdx.x * blockDim.x + threadIdx.x;
    if (i < n) dis[i] = rsqrtf((float)(counts[i] + 1));   // +1 self loop, deg >= 1
}

__global__ void fill_kernel(const int* __restrict__ edges, int* __restrict__ cursor,
                            int* __restrict__ csr)
{
    int i = blockIdx.x * blockDim.x + threadIdx.x;
    if (i < NEDGES) {
        int s = edges[2 * i];
        int d = edges[2 * i + 1];
        int slot = atomicAdd(&cursor[d], 1);
        csr[slot] = s;
    }
}

// ---------------------------------------------------------------------------
// Gather aggregation: one wave per node, 4 features per lane (float4 rows).
// out[t] = relu( dis[t] * sum_{s in in(t)} dis[s]*H[s] + dis[t]^2 * H[t] + b )
// ---------------------------------------------------------------------------
__global__ __launch_bounds__(256)
void agg_kernel(const float* __restrict__ H, const int* __restrict__ csr,
                const int* __restrict__ offs, const float* __restrict__ dis,
                const float* __restrict__ bias, float* __restrict__ OUT, int n)
{
    int node = blockIdx.x * 8 + (threadIdx.x >> 5);
    if (node >= n) return;
    int lane = threadIdx.x & 31;
    const float4* H4 = (const float4*)H;   // 32 float4 per row

    float4 acc = make_float4(0.f, 0.f, 0.f, 0.f);
    int e0 = offs[node], e1 = offs[node + 1];
    for (int e = e0; e < e1; ++e) {
        int s = csr[e];
        float ds = dis[s];
        float4 h = H4[(size_t)s * 32 + lane];
        acc.x += ds * h.x; acc.y += ds * h.y;
        acc.z += ds * h.z; acc.w += ds * h.w;
    }
    float dt  = dis[node];
    float dt2 = dt * dt;
    float4 hs = H4[(size_t)node * 32 + lane];
    float4 r;
    r.x = dt * acc.x + dt2 * hs.x + bias[lane * 4 + 0];
    r.y = dt * acc.y + dt2 * hs.y + bias[lane * 4 + 1];
    r.z = dt * acc.z + dt2 * hs.z + bias[lane * 4 + 2];
    r.w = dt * acc.w + dt2 * hs.w + bias[lane * 4 + 3];
    r.x = r.x > 0.f ? r.x : 0.f;
    r.y = r.y > 0.f ? r.y : 0.f;
    r.z = r.z > 0.f ? r.z : 0.f;
    r.w = r.w > 0.f ? r.w : 0.f;
    ((float4*)OUT)[(size_t)node * 32 + lane] = r;
}

// ---------------------------------------------------------------------------
// Deterministic two-stage mean over nodes, then the tiny actor/critic heads.
// ---------------------------------------------------------------------------
#define RCHUNK 196   // 196*256 = 50176 >= 50000

__global__ __launch_bounds__(128)
void reduce1_kernel(const float* __restrict__ X, float* __restrict__ partials, int n)
{
    int f = threadIdx.x;
    int b = blockIdx.x;
    int r0 = b * RCHUNK;
    int r1 = r0 + RCHUNK; if (r1 > n) r1 = n;
    float s = 0.f;
    for (int r = r0; r < r1; ++r) s += X[(size_t)r * DD + f];
    partials[b * DD + f] = s;
}

__global__ __launch_bounds__(128)
void reduce2_kernel(const float* __restrict__ partials, float* __restrict__ g)
{
    int f = threadIdx.x;
    float s = 0.f;
    for (int b = 0; b < 256; ++b) s += partials[b * DD + f];
    g[f] = s * (1.0f / (float)NNODES);
}

__global__ __launch_bounds__(32)
void head_kernel(const float* __restrict__ g,
                 const float* __restrict__ Wa, const float* __restrict__ ba,
                 const float* __restrict__ Wc, const float* __restrict__ bc,
                 float* __restrict__ out)
{
    int t = threadIdx.x;
    if (t < 16) {
        float acc = ba[t];
        for (int k = 0; k < DD; ++k) acc += g[k] * Wa[k * 16 + t];
        out[t] = acc;
    } else if (t == 16) {
        float acc = bc[0];
        for (int k = 0; k < DD; ++k) acc += g[k] * Wc[k];
        out[16] = acc;
    }
}

// ---------------------------------------------------------------------------
extern "C" void kernel_launch(void* const* d_in, const int* in_sizes, int n_in,
                              void* d_out, int out_size, void* d_ws, size_t ws_size,
                              hipStream_t stream)
{
    (void)in_sizes; (void)n_in; (void)out_size; (void)ws_size;

    const float* nodes = (const float*)d_in[0];
    const int*   edges = (const int*)  d_in[1];   // [E][2] int32 per harness
    const float* W_fc  = (const float*)d_in[2];
    const float* b_fc  = (const float*)d_in[3];
    const float* W1    = (const float*)d_in[4];
    const float* b1    = (const float*)d_in[5];
    const float* W2    = (const float*)d_in[6];
    const float* b2    = (const float*)d_in[7];
    const float* W_a   = (const float*)d_in[8];
    const float* b_a   = (const float*)d_in[9];
    const float* W_c   = (const float*)d_in[10];
    const float* b_c   = (const float*)d_in[11];
    float* out = (float*)d_out;

    // Workspace layout (all 4-byte types; base is 256B-aligned from hipMalloc)
    float* X1       = (float*)d_ws;                        // 50000*128
    float* H        = X1 + (size_t)NNODES * DD;            // 50000*128
    float* X2       = H  + (size_t)NNODES * DD;            // 50000*128
    float* dis      = X2 + (size_t)NNODES * DD;            // 50000
    float* partials = dis + NNODES;                        // 256*128
    float* g        = partials + 256 * DD;                 // 128
    int*   counts   = (int*)(g + DD);                      // 50000
    int*   offs     = counts + NNODES;                     // 50001
    int*   cursor   = offs + NNODES + 1;                   // 50000
    int*   csr      = cursor + NNODES;                     // 1600000

    // --- build CSR (dst-grouped) + degree normalizers (re-done every call) ---
    hipMemsetAsync(counts, 0, (size_t)NNODES * sizeof(int), stream);
    count_kernel<<<(NEDGES + 255) / 256, 256, 0, stream>>>(edges, counts);
    scan_kernel<<<1, 1024, 0, stream>>>(counts, offs, cursor, NNODES);
    dis_kernel<<<(NNODES + 255) / 256, 256, 0, stream>>>(counts, dis, NNODES);
    fill_kernel<<<(NEDGES + 255) / 256, 256, 0, stream>>>(edges, cursor, csr);

    // --- X1 = relu(nodes @ W_fc + b_fc)  (WMMA, K=256) ---
    dim3 ggrid(NNODES / 16, 2);
    gemm_wmma_kernel<<<ggrid, 128, 0, stream>>>(nodes, W_fc, b_fc, X1, NNODES, FIN, 1);

    // --- conv1: H = X1 @ W1 ; X2 = relu(aggregate(H) + b1) ---
    gemm_wmma_kernel<<<ggrid, 128, 0, stream>>>(X1, W1, nullptr, H, NNODES, DD, 0);
    agg_kernel<<<(NNODES + 7) / 8, 256, 0, stream>>>(H, csr, offs, dis, b1, X2, NNODES);

    // --- conv2: H = X2 @ W2 ; X1 = relu(aggregate(H) + b2) ---
    gemm_wmma_kernel<<<ggrid, 128, 0, stream>>>(X2, W2, nullptr, H, NNODES, DD, 0);
    agg_kernel<<<(NNODES + 7) / 8, 256, 0, stream>>>(H, csr, offs, dis, b2, X1, NNODES);

    // --- g = mean(X1, axis=0); heads ---
    reduce1_kernel<<<256, 128, 0, stream>>>(X1, partials, NNODES);
    reduce2_kernel<<<1, 128, 0, stream>>>(partials, g);
    head_kernel<<<1, 32, 0, stream>>>(g, W_a, b_a, W_c, b_c, out);
}